// Net_23587960389982
// MI455X (gfx1250) — compile-verified
//
#include <hip/hip_runtime.h>
#include <cstdint>
#include <math.h>

#define N_NODES 100000
#define N_EDGES 3200000
#define IN_DIM  1433
#define HID     16
#define OUT_DIM 7

typedef __attribute__((ext_vector_type(2))) float v2f;
typedef __attribute__((ext_vector_type(8))) float v8f;

// ---------------- degree / normalization ----------------

__global__ void k_deg_init(float* __restrict__ deg) {
    int i = blockIdx.x * blockDim.x + threadIdx.x;
    if (i < N_NODES) deg[i] = 1.0f;   // self loop contributes 1
}

__global__ void k_deg_edges(const int* __restrict__ row, float* __restrict__ deg) {
    int e = blockIdx.x * blockDim.x + threadIdx.x;
    if (e < N_EDGES) atomicAdd(&deg[row[e]], 1.0f);
}

__global__ void k_dinv(float* __restrict__ deg) {
    int i = blockIdx.x * blockDim.x + threadIdx.x;
    if (i < N_NODES) deg[i] = rsqrtf(deg[i]);   // deg >= 1 always (self loops)
}

// ---------------- GEMM1: h = x @ W1 via V_WMMA_F32_16X16X4_F32 ----------------
// One wave per 16-row node tile; HID==16 -> one 16x16 fp32 accumulator per tile.
// A 16x4 fp32 layout: lanes 0-15 hold M=lane, K={k,k+1}; lanes 16-31 hold K={k+2,k+3}.
// B 4x16 fp32 layout mirrors A: lane n<16 holds B[k..k+1][n]; lane n>=16 holds B[k+2..k+3][n-16].
// C/D 16x16 fp32: VGPR j, lanes 0-15 -> M=j, lanes 16-31 -> M=j+8.

__global__ void k_gemm1(const float* __restrict__ x, const float* __restrict__ W1,
                        float* __restrict__ h) {
    __shared__ float ldsW[IN_DIM * HID];          // 91,712 B  (< 320KB/WGP)
    for (int i = threadIdx.x; i < IN_DIM * HID; i += blockDim.x) ldsW[i] = W1[i];
    __syncthreads();

    const int wave = threadIdx.x >> 5;
    const int lane = threadIdx.x & 31;
    const int half = lane >> 4;       // 0: K lo pair, 1: K hi pair
    const int l16  = lane & 15;

    const int tile = blockIdx.x * 8 + wave;       // 8 waves / block
    if (tile >= N_NODES / 16) return;             // wave-uniform: EXEC stays all-1s
    const int m0 = tile * 16;

    const size_t rowbase = (size_t)(m0 + l16) * IN_DIM;

    v8f c = {0.f, 0.f, 0.f, 0.f, 0.f, 0.f, 0.f, 0.f};

    int k = 0;
    for (; k + 4 <= IN_DIM; k += 4) {
        const int ca = k + 2 * half;
        v2f a, b;
        a.x = x[rowbase + ca];
        a.y = x[rowbase + ca + 1];
        b.x = ldsW[ca * HID + l16];
        b.y = ldsW[(ca + 1) * HID + l16];
        c = __builtin_amdgcn_wmma_f32_16x16x4_f32(false, a, false, b,
                                                  (short)0, c, false, false);
    }
    // K tail: IN_DIM % 4 == 1 -> only column 1432 valid, zero-pad the rest
    {
        const int ca = k + 2 * half;
        v2f a = {0.f, 0.f}, b = {0.f, 0.f};
        if (ca < IN_DIM)     { a.x = x[rowbase + ca];     b.x = ldsW[ca * HID + l16]; }
        if (ca + 1 < IN_DIM) { a.y = x[rowbase + ca + 1]; b.y = ldsW[(ca + 1) * HID + l16]; }
        c = __builtin_amdgcn_wmma_f32_16x16x4_f32(false, a, false, b,
                                                  (short)0, c, false, false);
    }

#pragma unroll
    for (int j = 0; j < 8; ++j)
        h[(size_t)(m0 + j + 8 * half) * HID + l16] = c[j];
}

// ---------------- aggregation layer 1 ----------------

__global__ void k_agg1_init(const float* __restrict__ h, const float* __restrict__ dinv,
                            const float* __restrict__ b1, float* __restrict__ hagg) {
    int t = blockIdx.x * blockDim.x + threadIdx.x;
    if (t >= N_NODES * HID) return;
    int i = t >> 4, f = t & 15;
    float di = dinv[i];
    hagg[t] = b1[f] + di * di * h[t];             // bias + self-loop message
}

__global__ void k_agg1_edges(const int* __restrict__ row, const int* __restrict__ col,
                             const float* __restrict__ dinv, const float* __restrict__ h,
                             float* __restrict__ hagg) {
    int t = blockIdx.x * blockDim.x + threadIdx.x;
    if (t >= N_EDGES * 4) return;
    int e = t >> 2, q = t & 3;                    // 4 threads/edge, 4 feats each
    int r = row[e], cl = col[e];
    float nrm = dinv[r] * dinv[cl];
    const float4 hv = *(const float4*)(h + (size_t)cl * HID + q * 4); // 16B aligned
    float* dst = hagg + (size_t)r * HID + q * 4;
    atomicAdd(dst + 0, nrm * hv.x);
    atomicAdd(dst + 1, nrm * hv.y);
    atomicAdd(dst + 2, nrm * hv.z);
    atomicAdd(dst + 3, nrm * hv.w);
}

// ---------------- layer 2 dense: h2 = relu(hagg) @ W2 ----------------

__global__ void k_mlp2(const float* __restrict__ hagg, const float* __restrict__ W2,
                       float* __restrict__ h2) {
    __shared__ float w[HID * OUT_DIM];
    if (threadIdx.x < HID * OUT_DIM) w[threadIdx.x] = W2[threadIdx.x];
    __syncthreads();
    int i = blockIdx.x * blockDim.x + threadIdx.x;
    if (i >= N_NODES) return;
    float hv[HID];
    const float4* hp = (const float4*)(hagg + (size_t)i * HID);
#pragma unroll
    for (int q = 0; q < 4; ++q) {
        float4 v = hp[q];
        hv[q * 4 + 0] = fmaxf(v.x, 0.f);
        hv[q * 4 + 1] = fmaxf(v.y, 0.f);
        hv[q * 4 + 2] = fmaxf(v.z, 0.f);
        hv[q * 4 + 3] = fmaxf(v.w, 0.f);
    }
#pragma unroll
    for (int of = 0; of < OUT_DIM; ++of) {
        float acc = 0.f;
#pragma unroll
        for (int kk = 0; kk < HID; ++kk) acc = fmaf(hv[kk], w[kk * OUT_DIM + of], acc);
        h2[(size_t)i * OUT_DIM + of] = acc;
    }
}

// ---------------- aggregation layer 2 (into d_out) ----------------

__global__ void k_agg2_init(const float* __restrict__ h2, const float* __restrict__ dinv,
                            const float* __restrict__ b2, float* __restrict__ out) {
    int t = blockIdx.x * blockDim.x + threadIdx.x;
    if (t >= N_NODES * OUT_DIM) return;
    int i = t / OUT_DIM, f = t % OUT_DIM;
    float di = dinv[i];
    out[t] = b2[f] + di * di * h2[t];
}

__global__ void k_agg2_edges(const int* __restrict__ row, const int* __restrict__ col,
                             const float* __restrict__ dinv, const float* __restrict__ h2,
                             float* __restrict__ out) {
    int e = blockIdx.x * blockDim.x + threadIdx.x;
    if (e >= N_EDGES) return;
    int r = row[e], cl = col[e];
    float nrm = dinv[r] * dinv[cl];
    const float* src = h2 + (size_t)cl * OUT_DIM;
    float* dst = out + (size_t)r * OUT_DIM;
#pragma unroll
    for (int f = 0; f < OUT_DIM; ++f) atomicAdd(dst + f, nrm * src[f]);
}

// ---------------- log_softmax (in place on d_out) ----------------

__global__ void k_logsoftmax(float* __restrict__ out) {
    int i = blockIdx.x * blockDim.x + threadIdx.x;
    if (i >= N_NODES) return;
    float* p = out + (size_t)i * OUT_DIM;
    float v[OUT_DIM];
    float m = -INFINITY;
#pragma unroll
    for (int f = 0; f < OUT_DIM; ++f) { v[f] = p[f]; m = fmaxf(m, v[f]); }
    float s = 0.f;
#pragma unroll
    for (int f = 0; f < OUT_DIM; ++f) s += expf(v[f] - m);
    float ls = logf(s);
#pragma unroll
    for (int f = 0; f < OUT_DIM; ++f) p[f] = v[f] - m - ls;
}

// ---------------- launch ----------------

extern "C" void kernel_launch(void* const* d_in, const int* in_sizes, int n_in,
                              void* d_out, int out_size, void* d_ws, size_t ws_size,
                              hipStream_t stream) {
    const float* x  = (const float*)d_in[0];
    const int*   ei = (const int*)d_in[1];      // [2, E] flat: row then col
    const float* W1 = (const float*)d_in[2];
    const float* b1 = (const float*)d_in[3];
    const float* W2 = (const float*)d_in[4];
    const float* b2 = (const float*)d_in[5];
    const int* row = ei;
    const int* col = ei + N_EDGES;
    float* out = (float*)d_out;

    // workspace: dinv[N] | h[N*16] | hagg[N*16] | h2[N*7]  = 16 MB total
    float* ws   = (float*)d_ws;
    float* dinv = ws;
    float* h    = dinv + N_NODES;
    float* hagg = h + (size_t)N_NODES * HID;
    float* h2   = hagg + (size_t)N_NODES * HID;

    dim3 B(256);
    k_deg_init  <<<(N_NODES + 255) / 256, B, 0, stream>>>(dinv);
    k_deg_edges <<<(N_EDGES + 255) / 256, B, 0, stream>>>(row, dinv);
    k_dinv      <<<(N_NODES + 255) / 256, B, 0, stream>>>(dinv);

    int tiles = N_NODES / 16;                       // 6250
    k_gemm1     <<<(tiles + 7) / 8, B, 0, stream>>>(x, W1, h);

    k_agg1_init <<<(N_NODES * HID + 255) / 256, B, 0, stream>>>(h, dinv, b1, hagg);
    k_agg1_edges<<<(N_EDGES * 4 + 255) / 256, B, 0, stream>>>(row, col, dinv, h, hagg);

    k_mlp2      <<<(N_NODES + 255) / 256, B, 0, stream>>>(hagg, W2, h2);

    k_agg2_init <<<(N_NODES * OUT_DIM + 255) / 256, B, 0, stream>>>(h2, dinv, b2, out);
    k_agg2_edges<<<(N_EDGES + 255) / 256, B, 0, stream>>>(row, col, dinv, h2, out);

    k_logsoftmax<<<(N_NODES + 255) / 256, B, 0, stream>>>(out);
}